// _Switch_8847632630028
// MI455X (gfx1250) — compile-verified
//
#include <hip/hip_runtime.h>

// MI455X / gfx1250, wave32. Switch-MoE 2-layer MLP, fp32 end-to-end using
// V_WMMA_F32_16X16X4_F32 (native fp32 matrix path on CDNA5).

typedef __attribute__((ext_vector_type(2))) float v2f;
typedef __attribute__((ext_vector_type(8))) float v8f;

#define NMOD 8
#define HID 2048
#define DOUT 4096
#define FPM 512
#define NB 4096
#define MAXPAD (NB + NMOD * 16)   // 4224: worst-case padded row count
#define MAXTILES (MAXPAD / 16)    // 264 row tiles of 16
#define KC 32                     // K-chunk staged in LDS
#define LDSTRIDE 272              // 256 + 16 pad: lane halves (k vs k+2) hit disjoint banks
#define DBLK 256                  // D columns per block (8 waves x 32 cols)

// ---------------------------------------------------------------------------
// Kernel 1: stable grouping of samples by expert, one wave32.
// All lanes keep identical counts/offsets in registers via ballot+popcount,
// so no LDS and no barriers are needed; ranks make the scatter stable.
// ---------------------------------------------------------------------------
__global__ __launch_bounds__(32) void group_kernel(const float* __restrict__ batch_in,
                                                   int* __restrict__ perm,
                                                   int* __restrict__ tileMod)
{
    const int lane = threadIdx.x;
    const unsigned lt = (1u << lane) - 1u;

    int cnt[NMOD];
#pragma unroll
    for (int e = 0; e < NMOD; ++e) cnt[e] = 0;

    for (int i0 = 0; i0 < NB; i0 += 32) {
        int f = (int)batch_in[i0 + lane];
        int m = f >> 9;                       // / FPM
#pragma unroll
        for (int e = 0; e < NMOD; ++e)
            cnt[e] += __popc((unsigned)__ballot(m == e));
    }

    int off[NMOD + 1];
    off[0] = 0;
#pragma unroll
    for (int e = 0; e < NMOD; ++e) off[e + 1] = off[e] + ((cnt[e] + 15) & ~15);

    int run[NMOD];
#pragma unroll
    for (int e = 0; e < NMOD; ++e) run[e] = off[e];

    for (int i0 = 0; i0 < NB; i0 += 32) {
        int f = (int)batch_in[i0 + lane];
        int m = f >> 9;
#pragma unroll
        for (int e = 0; e < NMOD; ++e) {
            unsigned bal = (unsigned)__ballot(m == e);
            if (m == e) perm[run[e] + __popc(bal & lt)] = i0 + lane;
            run[e] += __popc(bal);            // uniform across lanes
        }
    }

    // pad group tails and the region beyond the last real tile with -1
#pragma unroll
    for (int e = 0; e < NMOD; ++e)
        for (int p = run[e] + lane; p < off[e + 1]; p += 32) perm[p] = -1;
    for (int p = off[NMOD] + lane; p < MAXPAD; p += 32) perm[p] = -1;

    // tile -> expert map (tiles past the end map to expert 0; all rows -1)
    for (int t = lane; t < MAXTILES; t += 32) {
        int r = t * 16, e = 0;
#pragma unroll
        for (int q = 0; q < NMOD; ++q)
            if (r >= off[q] && r < off[q + 1]) e = q;
        tileMod[t] = e;
    }
}

// ---------------------------------------------------------------------------
// Kernel 2: fused layer1+layer2 GEMM. Block = 8 waves = 16 rows x 256 cols.
// Wave = 16x32 (two 16x16 f32 accumulators). A-frag regenerated on the fly.
// ---------------------------------------------------------------------------
__global__ __launch_bounds__(256) void moe_wmma_kernel(
    const float* __restrict__ batch_in,
    const float* __restrict__ W1, const float* __restrict__ b1,
    const float* __restrict__ W2, const float* __restrict__ b2,
    const int* __restrict__ perm, const int* __restrict__ tileMod,
    float* __restrict__ out)
{
    __shared__ float ldsW2[KC * LDSTRIDE];   // ~34.8 KB
    __shared__ float ldsW1[KC];
    __shared__ float ldsB1[KC];
    __shared__ int   permS[16];

    const int tile = blockIdx.x;             // row-tile major: L2 reuse of W2 slice
    const int d0   = blockIdx.y * DBLK;
    const int tid  = threadIdx.x;
    const int lane = tid & 31;
    const int wv   = tid >> 5;
    const int m    = tileMod[tile];

    if (tid < 16) permS[tid] = perm[tile * 16 + tid];
    __syncthreads();

    // A-fragment geometry for V_WMMA_F32_16X16X4_F32 (16x4, 2 VGPRs):
    // lanes 0-15 hold rows 0-15 at K = {k0, k0+1}; lanes 16-31 at K = {k0+2, k0+3}.
    const int   row   = lane & 15;
    const int   halfK = (lane >> 4) << 1;
    const int   smp   = permS[row];
    const float fin   = (smp >= 0) ? (float)(((int)batch_in[smp]) & (FPM - 1)) : 0.0f;

    const float* __restrict__ W2m = W2 + (size_t)m * HID * DOUT + d0;
    const float* __restrict__ W1m = W1 + m * HID;
    const float* __restrict__ b1m = b1 + m * HID;

    const int colW = (wv << 5) + (lane & 15);   // this wave's column within block tile

    v8f acc0 = {};
    v8f acc1 = {};

    for (int kc0 = 0; kc0 < HID; kc0 += KC) {
        __syncthreads();   // WAR: previous chunk fully consumed
        {
            const float* src = W2m + (size_t)kc0 * DOUT;
#pragma unroll
            for (int it = 0; it < (KC * DBLK / 4) / 256; ++it) {   // 8 float4 / thread
                int idx = tid + it * 256;
                int kk  = idx >> 6;            // / (DBLK/4)
                int dd  = (idx & 63) << 2;
                float4 v = *(const float4*)(src + (size_t)kk * DOUT + dd);
                *(float4*)(&ldsW2[kk * LDSTRIDE + dd]) = v;        // ds_store_b128
                if (kc0 + KC < HID)            // pull next K-chunk toward L2/L0
                    __builtin_prefetch(src + (size_t)(kk + KC) * DOUT + dd, 0, 3);
            }
            if (tid < KC)            ldsW1[tid]      = W1m[kc0 + tid];
            else if (tid < 2 * KC)   ldsB1[tid - KC] = b1m[kc0 + tid - KC];
        }
        __syncthreads();

#pragma unroll
        for (int kk = 0; kk < KC; kk += 4) {
            const int kr = kk + halfK;
            v2f a;   // h = relu(fin * W1 + b1), regenerated per K-step
            a.x = fmaxf(fmaf(fin, ldsW1[kr],     ldsB1[kr]),     0.0f);
            a.y = fmaxf(fmaf(fin, ldsW1[kr + 1], ldsB1[kr + 1]), 0.0f);
            v2f bA, bB;  // B-frag (4x16): same K split across lane halves as A
            bA.x = ldsW2[kr * LDSTRIDE + colW];
            bA.y = ldsW2[(kr + 1) * LDSTRIDE + colW];
            bB.x = ldsW2[kr * LDSTRIDE + colW + 16];
            bB.y = ldsW2[(kr + 1) * LDSTRIDE + colW + 16];
            acc0 = __builtin_amdgcn_wmma_f32_16x16x4_f32(false, a, false, bA,
                                                         (short)0, acc0, false, false);
            acc1 = __builtin_amdgcn_wmma_f32_16x16x4_f32(false, a, false, bB,
                                                         (short)0, acc1, false, false);
        }
    }

    // Epilogue. C/D layout: VGPR j -> row j (lanes 0-15) / row j+8 (lanes 16-31),
    // col = lane & 15. Add b2 and scatter rows back via perm; skip padding rows.
    const int c0   = d0 + (wv << 5) + (lane & 15);
    const float b2v0 = b2[(size_t)m * DOUT + c0];
    const float b2v1 = b2[(size_t)m * DOUT + c0 + 16];
    const int rbase = (lane >> 4) << 3;
#pragma unroll
    for (int j = 0; j < 8; ++j) {
        int s = permS[rbase + j];
        if (s >= 0) {
            out[(size_t)s * DOUT + c0]      = acc0[j] + b2v0;
            out[(size_t)s * DOUT + c0 + 16] = acc1[j] + b2v1;
        }
    }
}

// ---------------------------------------------------------------------------
extern "C" void kernel_launch(void* const* d_in, const int* in_sizes, int n_in,
                              void* d_out, int out_size, void* d_ws, size_t ws_size,
                              hipStream_t stream)
{
    const float* batch_in = (const float*)d_in[0];   // [B,1] float frame ids
    const float* W1       = (const float*)d_in[1];   // [M,1,H]
    const float* b1       = (const float*)d_in[2];   // [M,H]
    const float* W2       = (const float*)d_in[3];   // [M,H,D]
    const float* b2       = (const float*)d_in[4];   // [M,D]
    // d_in[5] = fpm (== 512), folded into compile-time constants
    (void)in_sizes; (void)n_in; (void)out_size; (void)ws_size;

    int* perm    = (int*)d_ws;          // MAXPAD ints, fully rewritten each call
    int* tileMod = perm + MAXPAD;       // MAXTILES ints

    group_kernel<<<1, 32, 0, stream>>>(batch_in, perm, tileMod);
    moe_wmma_kernel<<<dim3(MAXTILES, DOUT / DBLK), 256, 0, stream>>>(
        batch_in, W1, b1, W2, b2, perm, tileMod, (float*)d_out);
}